// SpatialGATLayer_69123203662646
// MI455X (gfx1250) — compile-verified
//
#include <hip/hip_runtime.h>
#include <math.h>
#include <stdint.h>

typedef __attribute__((ext_vector_type(2))) float v2f;
typedef __attribute__((ext_vector_type(8))) float v8f;
typedef __attribute__((ext_vector_type(4))) unsigned int u32x4;
typedef __attribute__((ext_vector_type(8))) int i32x8;
typedef __attribute__((ext_vector_type(4))) int i32x4;

#define BB 4
#define NN 128
#define TT 96
#define DH 128
#define NHEADS 8
#define DK 16
#define GAT_NEG_INF -1000000000.0f
#define APAD 132   // padded row stride (floats): 4-bank skew per row, 16B aligned

__device__ __forceinline__ v8f wmma_f32_16x16x4(v2f a, v2f b, v8f c) {
    // 8 args: (neg_a, A, neg_b, B, c_mod, C, reuse_a, reuse_b)
    return __builtin_amdgcn_wmma_f32_16x16x4_f32(false, a, false, b, (short)0, c,
                                                 false, false);
}

// Issue a 2D TENSOR_LOAD_TO_LDS (TDM DMA). All values wave-uniform -> SGPRs.
//  dim0 = contiguous length (elements), dim1 = #rows, stride0 in elements.
//  padw0 ORs the pad_enable/pad_interval/pad_amount bits into group1 word0.
__device__ __forceinline__ void tdm_load_2d_f32(uint32_t lds_byte_addr,
                                                uint64_t global_byte_addr,
                                                uint32_t dim0, uint32_t dim1,
                                                uint32_t stride0, uint32_t padw0) {
    u32x4 g0 = {
        1u,                                            // count=1, user descriptor
        lds_byte_addr,                                 // LDS destination (bytes)
        (uint32_t)global_byte_addr,                    // global addr lo
        (uint32_t)(global_byte_addr >> 32) | (2u << 30)// global addr hi | type=2
    };
    i32x8 g1 = {
        (int)((2u << 16) | padw0),   // data_size=4B (+optional pad config)
        (int)(dim0 << 16),           // tensor_dim0[15:0]
        (int)(dim1 << 16),           // tensor_dim0 hi=0 | tensor_dim1[15:0]
        (int)(dim0 << 16),           // tensor_dim1 hi=0 | tile_dim0
        (int)dim1,                   // tile_dim1 | tile_dim2=0
        (int)stride0,                // tensor_dim0_stride lo
        0, 0                         // stride hi / dim1_stride (unused, 2D tile)
    };
    i32x4 g2 = {0, 0, 0, 0};
    i32x4 g3 = {0, 0, 0, 0};
    i32x8 g4 = {0, 0, 0, 0, 0, 0, 0, 0};   // extra group, zero-filled (clang-23 form)
    __builtin_amdgcn_tensor_load_to_lds(g0, g1, g2, g3, g4, 0);
}

// ---------------------------------------------------------------------------
// Kernel 1: Wh[row, e] = sum_d Z[row, d] * W[e, d]   (row = flattened b,n,t)
// Each WG: 16 rows x 128 cols. Wave w handles head w (cols 16w..16w+15).
// Z tile staged by TDM with LDS padding (132-float rows -> conflict-free
// ds_load_2addr_b64 A-fragments). Epilogue stores Wh and reduces
// s1 = Wh_head.a1, s2 = Wh_head.a2 via shfl trees (wave's cols == one head).
// ---------------------------------------------------------------------------
__global__ __launch_bounds__(256) void gat_gemm1_kernel(
    const float* __restrict__ Z, const float* __restrict__ W,
    const float* __restrict__ a, float* __restrict__ Wh,
    float* __restrict__ s1, float* __restrict__ s2) {
    __shared__ float zs[16 * APAD];

    const int tid = threadIdx.x;
    const int rowbase = blockIdx.x * 16;

    if (tid < 32) {
        // 16 rows x 128 floats, pad 4 DWORDs (pad_amount=3) every 128 DWORDs
        // (pad_interval=6: 512B) -> effective LDS row stride = APAD floats.
        const uint32_t padw0 = (1u << 20) | (6u << 22) | (3u << 25);
        tdm_load_2d_f32((uint32_t)(uintptr_t)zs,
                        (uint64_t)(uintptr_t)(Z + (size_t)rowbase * DH),
                        /*dim0=*/DH, /*dim1=*/16, /*stride0=*/DH, padw0);
        __builtin_amdgcn_s_wait_tensorcnt(0);
    }
    __syncthreads();

    const int wave = tid >> 5;       // head 0..7
    const int lane = tid & 31;
    const int n    = lane & 15;      // output col within head / frag lane
    const int half = lane >> 4;      // K sub-offset selector
    const int ebase = wave * 16;

    v8f acc = {0.f, 0.f, 0.f, 0.f, 0.f, 0.f, 0.f, 0.f};
    const float* wrow = W + (size_t)(ebase + n) * DH;  // B[d][e] = W[e][d]
    const float* zrow = &zs[n * APAD];

#pragma unroll
    for (int kk = 0; kk < 32; ++kk) {
        int k0 = kk * 4 + half * 2;
        v2f afrag = *(const v2f*)(zrow + k0);     // Z[row=n][k0, k0+1]
        v2f bfrag = *(const v2f*)(wrow + k0);     // W[e][k0, k0+1]
        acc = wmma_f32_16x16x4(afrag, bfrag, acc);
    }

    // Epilogue: store Wh and reduce s1/s2 per row.
    const float a1v = a[n];
    const float a2v = a[DK + n];
#pragma unroll
    for (int r = 0; r < 8; ++r) {
        int row = rowbase + r + half * 8;          // D vgpr r -> rows r / r+8
        float d = acc[r];
        Wh[(size_t)row * DH + ebase + n] = d;
        float t1 = d * a1v;
        float t2 = d * a2v;
#pragma unroll
        for (int m = 1; m <= 8; m <<= 1) {         // reduce within 16-lane half
            t1 += __shfl_xor(t1, m, 32);
            t2 += __shfl_xor(t2, m, 32);
        }
        if (n == 0) {
            int sidx = row * NHEADS + wave;
            s1[sidx] = t1;
            s2[sidx] = t2;
        }
    }
}

// ---------------------------------------------------------------------------
// Kernel 2: one WG per (b, t, h).
//  - V tile (128 j-rows x 16 k) DMA'd into LDS by TDM; its latency hides
//    behind the whole softmax phase (tensorcnt waited only at the pre-GEMM
//    barrier).
//  - softmax rows: wave w owns rows 16w..16w+15; 2 lanes/row, 64 j each.
//  - alpha (exp values, un-normalized) in padded LDS tile; 1/sum and ELU
//    folded into the WMMA epilogue.
//  B-fragment banks: lanes 0-15 read banks n+{0,16}, lanes 16-31 banks
//  n+{32,48} (k0 multiple of 4, row = 16 floats) -> conflict-free.
// ---------------------------------------------------------------------------
__global__ __launch_bounds__(256) void gat_attn_kernel(
    const float* __restrict__ Wh, const float* __restrict__ A,
    const float* __restrict__ s1, const float* __restrict__ s2,
    float* __restrict__ out) {
    extern __shared__ float smem[];
    float* alpha    = smem;                         // 128 * APAD
    float* Vt       = alpha + NN * APAD;            // 128 * 16   V[j][k]
    float* sjs      = Vt + NN * DK;                 // 128
    float* rowscale = sjs + NN;                     // 128

    const int tid = threadIdx.x;
    const int h = blockIdx.x & 7;
    const int t = (blockIdx.x >> 3) % TT;
    const int b = blockIdx.x / (8 * TT);

    // Kick off V DMA: V[j][k] = Wh[((b*N + j)*T + t)*DH + h*16 + k]
    const float* Vbase = Wh + ((size_t)(b * NN) * TT + t) * DH + h * DK;
    if (tid < 32) {
        tdm_load_2d_f32((uint32_t)(uintptr_t)Vt,
                        (uint64_t)(uintptr_t)Vbase,
                        /*dim0=*/DK, /*dim1=*/NN, /*stride0=*/TT * DH, 0u);
    }
    if (tid < NN) {
        sjs[tid] = s2[((b * NN + tid) * TT + t) * NHEADS + h];
    }
    __syncthreads();   // sjs ready (V still in flight)

    const int wave = tid >> 5;
    const int lane = tid & 31;
    const int half = lane >> 4;                  // which 64-j slab
    const int i    = wave * 16 + (lane & 15);    // attention row 0..127

    const float si = s1[((b * NN + i) * TT + t) * NHEADS + h];
    const float* Arow = A + (size_t)i * NN + half * 64;
    float* arow = alpha + i * APAD + half * 64;

    // Pass 1: scores -> LDS, track max.
    float m = -INFINITY;
#pragma unroll 4
    for (int jj = 0; jj < 64; ++jj) {
        float Av = Arow[jj];
        int j = half * 64 + jj;
        float e = si + sjs[j];
        e = (e >= 0.f) ? e : 0.2f * e;                 // leaky_relu(0.2)
        bool msk = (Av > 0.f) || (i == j);             // (A>0)+eye clipped
        e = msk ? e : GAT_NEG_INF;
        e += 0.1f * Av;
        arow[jj] = e;
        m = fmaxf(m, e);
    }
    m = fmaxf(m, __shfl_xor(m, 16, 32));

    // Pass 2: exp in place, accumulate sum.
    float sum = 0.f;
#pragma unroll 4
    for (int jj = 0; jj < 64; ++jj) {
        float p = __expf(arow[jj] - m);
        arow[jj] = p;
        sum += p;
    }
    sum += __shfl_xor(sum, 16, 32);
    if (half == 0) rowscale[i] = 1.0f / sum;

    if (tid < 32) __builtin_amdgcn_s_wait_tensorcnt(0);  // V DMA done
    __syncthreads();                                     // alpha + V visible

    // GEMM: h_new rows 16w..16w+15 = alpha(16x128) @ V(128x16) via 32 WMMAs.
    const int n = lane & 15;                         // output col (k in head)
    v8f acc = {0.f, 0.f, 0.f, 0.f, 0.f, 0.f, 0.f, 0.f};
    const float* arows = alpha + (wave * 16 + n) * APAD;  // A-frag row
    const float* vcol  = Vt + n;                          // V[.][n]
#pragma unroll
    for (int kk = 0; kk < 32; ++kk) {
        int k0 = kk * 4 + half * 2;
        v2f af = *(const v2f*)(arows + k0);   // alpha[row][k0, k0+1]
        v2f bf;
        bf.x = vcol[(k0 + 0) * DK];           // V[k0][n]
        bf.y = vcol[(k0 + 1) * DK];           // V[k0+1][n]
        acc = wmma_f32_16x16x4(af, bf, acc);
    }

    // Epilogue: normalize, ELU, store (contiguous 64B per row segment).
    float* obase = out + ((size_t)(b * NN) * TT + t) * DH + h * DK + n;
#pragma unroll
    for (int r = 0; r < 8; ++r) {
        int row = wave * 16 + r + half * 8;
        float v = acc[r] * rowscale[row];
        v = (v > 0.f) ? v : (__expf(v) - 1.0f);       // elu
        obase[(size_t)row * TT * DH] = v;
    }
}

// ---------------------------------------------------------------------------
extern "C" void kernel_launch(void* const* d_in, const int* in_sizes, int n_in,
                              void* d_out, int out_size, void* d_ws, size_t ws_size,
                              hipStream_t stream) {
    const float* Z = (const float*)d_in[0];   // (B,N,T,d_h)
    const float* A = (const float*)d_in[1];   // (N,N)
    const float* W = (const float*)d_in[2];   // (d_h,d_h)
    const float* a = (const float*)d_in[3];   // (2*d_k)
    float* out = (float*)d_out;

    float* Wh = (float*)d_ws;                             // 25.2 MB
    float* s1 = Wh + (size_t)BB * NN * TT * DH;           // 1.5 MB
    float* s2 = s1 + (size_t)BB * NN * TT * NHEADS;       // 1.5 MB

    const int mtiles = (BB * NN * TT) / 16;               // 3072
    gat_gemm1_kernel<<<mtiles, 256, 0, stream>>>(Z, W, a, Wh, s1, s2);

    const int nslices = BB * TT * NHEADS;                 // 3072
    const size_t lds2 = (size_t)(NN * APAD + NN * DK + NN + NN) * sizeof(float);
    gat_attn_kernel<<<nslices, 256, lds2, stream>>>(Wh, A, s1, s2, out);
}